// GraphModel_23261542875812
// MI455X (gfx1250) — compile-verified
//
#include <hip/hip_runtime.h>
#include <hip/hip_bf16.h>

typedef __attribute__((ext_vector_type(2))) float v2f;
typedef __attribute__((ext_vector_type(8))) float v8f;

#define FDIM 64

// ---------------- degree / norm kernels ----------------

__global__ void deg_init_k(float* __restrict__ deg, int n) {
    int i = blockIdx.x * blockDim.x + threadIdx.x;
    if (i < n) deg[i] = 1.0f;              // self-loop contributes 1
}

__global__ void deg_acc_k(const int* __restrict__ dst, float* __restrict__ deg, int e) {
    int i = blockIdx.x * blockDim.x + threadIdx.x;
    if (i < e) atomicAdd(&deg[dst[i]], 1.0f);
}

__global__ void rsqrt_k(float* __restrict__ deg, int n) {
    int i = blockIdx.x * blockDim.x + threadIdx.x;
    if (i < n) deg[i] = rsqrtf(deg[i]);    // deg >= 1 always
}

// ---------------- WMMA GEMM: H[n,64] = X[n,64] @ W[64,64] ----------------
// One wave -> 16 rows x 64 cols. 256 threads = 8 waves -> 128 rows/block.
// K=64 consumed in 16 steps of V_WMMA_F32_16X16X4_F32 across 4 N-tiles.
// W is staged in LDS packed as K-pairs: sW[p*128 + n*2 + {0,1}] = {W[2p][n], W[2p+1][n]}
// so every B fragment is one aligned ds_load_b64 straight into the VGPR pair.

__global__ __launch_bounds__(256) void gemm64_wmma_k(const float* __restrict__ X,
                                                     const float* __restrict__ W,
                                                     float* __restrict__ H, int n) {
    __shared__ float sW[FDIM * FDIM];      // 16 KB, packed K-pair layout
    for (int i = threadIdx.x; i < (FDIM / 2) * FDIM; i += 256) {
        int p  = i >> 6;                   // K-pair index 0..31
        int nn = i & 63;                   // output column
        sW[(p * FDIM + nn) * 2 + 0] = W[(2 * p + 0) * FDIM + nn];
        sW[(p * FDIM + nn) * 2 + 1] = W[(2 * p + 1) * FDIM + nn];
    }
    __syncthreads();

    const int wave  = threadIdx.x >> 5;
    const int lane  = threadIdx.x & 31;
    const int m     = lane & 15;           // row within 16-row tile / B column within tile
    const int khalf = lane >> 4;           // 0: K={0,1}, 1: K={2,3}

    long long row0 = (long long)blockIdx.x * 128 + (long long)wave * 16;
    long long rowA = row0 + m;
    if (rowA > (long long)n - 1) rowA = (long long)n - 1;   // clamp reads: keep EXEC all-1s for WMMA
    const float* __restrict__ xrow = X + rowA * FDIM;

    v8f acc0 = {}, acc1 = {}, acc2 = {}, acc3 = {};

    #pragma unroll
    for (int k0 = 0; k0 < FDIM; k0 += 4) {
        const int ka = k0 + khalf * 2;     // even -> 8B-aligned v2f load
        v2f a = *(const v2f*)&xrow[ka];
        const int p = (k0 >> 1) + khalf;   // K-pair row in packed LDS
        const float* wrow = &sW[(p * FDIM + m) * 2];
        v2f b;
        b = *(const v2f*)&wrow[0];
        acc0 = __builtin_amdgcn_wmma_f32_16x16x4_f32(false, a, false, b, (short)0, acc0, false, false);
        b = *(const v2f*)&wrow[32];        // col + 16
        acc1 = __builtin_amdgcn_wmma_f32_16x16x4_f32(false, a, false, b, (short)0, acc1, false, false);
        b = *(const v2f*)&wrow[64];        // col + 32
        acc2 = __builtin_amdgcn_wmma_f32_16x16x4_f32(false, a, false, b, (short)0, acc2, false, false);
        b = *(const v2f*)&wrow[96];        // col + 48
        acc3 = __builtin_amdgcn_wmma_f32_16x16x4_f32(false, a, false, b, (short)0, acc3, false, false);
    }

    // C/D layout: VGPR j -> row (j + 8*khalf), col = m within each 16-col tile
    #pragma unroll
    for (int j = 0; j < 8; ++j) {
        long long r = row0 + j + khalf * 8;
        if (r < n) {
            float* hr = H + r * FDIM + m;
            hr[0]  = acc0[j];
            hr[16] = acc1[j];
            hr[32] = acc2[j];
            hr[48] = acc3[j];
        }
    }
}

// ---------------- aggregation ----------------

// out[i,f] = h[i,f]*dinv[i]^2 + bias[f]   (self-loop term + bias)
__global__ void selfloop_bias_k(const float* __restrict__ h, const float* __restrict__ dinv,
                                const float* __restrict__ bias, float* __restrict__ out,
                                long long total) {
    long long idx = (long long)blockIdx.x * blockDim.x + threadIdx.x;
    if (idx < total) {
        int node = (int)(idx >> 6);
        int f    = (int)(idx & 63);
        float d  = dinv[node];
        out[idx] = h[idx] * d * d + bias[f];
    }
}

// one wave per edge; lane l handles features l and l+32 (wave-contiguous atomics)
__global__ __launch_bounds__(256) void edge_scatter_k(const int* __restrict__ src,
                                                      const int* __restrict__ dst,
                                                      const float* __restrict__ dinv,
                                                      const float* __restrict__ h,
                                                      float* __restrict__ out, int e) {
    int edge = (int)(((long long)blockIdx.x * blockDim.x + threadIdx.x) >> 5);
    int lane = threadIdx.x & 31;
    if (edge >= e) return;
    int s = src[edge];
    int d = dst[edge];
    float nrm = dinv[s] * dinv[d];
    const float* hs = h   + (long long)s * FDIM;
    float*       od = out + (long long)d * FDIM;
    atomicAdd(&od[lane],      hs[lane]      * nrm);
    atomicAdd(&od[lane + 32], hs[lane + 32] * nrm);
}

__global__ void relu_k(float* __restrict__ x, long long total) {
    long long idx = (long long)blockIdx.x * blockDim.x + threadIdx.x;
    if (idx < total) x[idx] = fmaxf(x[idx], 0.0f);
}

// ---------------- host-side orchestration ----------------

extern "C" void kernel_launch(void* const* d_in, const int* in_sizes, int n_in,
                              void* d_out, int out_size, void* d_ws, size_t ws_size,
                              hipStream_t stream) {
    const float* embedding = (const float*)d_in[0];
    const int*   edge_idx  = (const int*)d_in[1];   // [2,E] flat: row0=src, row1=dst
    const float* W1        = (const float*)d_in[2];
    const float* b1        = (const float*)d_in[3];
    const float* W2        = (const float*)d_in[4];
    const float* b2        = (const float*)d_in[5];

    const int n = in_sizes[0] / FDIM;
    const int e = in_sizes[1] / 2;
    const int* src = edge_idx;
    const int* dst = edge_idx + e;

    // workspace partition
    long long n_pad = ((long long)n + 127) & ~127LL;
    float* dinv = (float*)d_ws;                    // n floats (deg, then rsqrt in place)
    float* hbuf = dinv + n_pad;                    // n*64: h = x@W of current layer
    float* h1   = hbuf + (long long)n * FDIM;      // n*64: layer-1 output
    float* outf = (float*)d_out;

    const long long total = (long long)n * FDIM;
    const int blkN   = (n + 255) / 256;
    const int blkE   = (e + 255) / 256;
    const int blkT   = (int)((total + 255) / 256);
    const int blkG   = (n + 127) / 128;            // gemm: 128 rows/block
    const int blkEs  = (int)(((long long)e * 32 + 255) / 256);

    // degrees -> dinv
    deg_init_k<<<blkN, 256, 0, stream>>>(dinv, n);
    deg_acc_k<<<blkE, 256, 0, stream>>>(dst, dinv, e);
    rsqrt_k<<<blkN, 256, 0, stream>>>(dinv, n);

    // ---- layer 1 ----
    gemm64_wmma_k<<<blkG, 256, 0, stream>>>(embedding, W1, hbuf, n);
    selfloop_bias_k<<<blkT, 256, 0, stream>>>(hbuf, dinv, b1, h1, total);
    edge_scatter_k<<<blkEs, 256, 0, stream>>>(src, dst, dinv, hbuf, h1, e);
    relu_k<<<blkT, 256, 0, stream>>>(h1, total);

    // ---- layer 2 ----
    gemm64_wmma_k<<<blkG, 256, 0, stream>>>(h1, W2, hbuf, n);
    selfloop_bias_k<<<blkT, 256, 0, stream>>>(hbuf, dinv, b2, outf, total);
    edge_scatter_k<<<blkEs, 256, 0, stream>>>(src, dst, dinv, hbuf, outf, e);
    relu_k<<<blkT, 256, 0, stream>>>(outf, total);
}